// SlaterDeterminant_54932631716070
// MI455X (gfx1250) — compile-verified
//
#include <hip/hip_runtime.h>
#include <math.h>

// CDNA5 / gfx1250: wave32, WMMA f32 16x16x4 for the LU trailing updates.

typedef __attribute__((ext_vector_type(2))) float v2f;
typedef __attribute__((ext_vector_type(8))) float v8f;

#define N 128
#define LDW 132              // 128 + 4 floats pad to break LDS bank conflicts
#define NBATCH 2048
#define PI_F 3.14159265358979f
#define TWO_PI_F 6.28318530717959f

// One block per matrix (blocks 0..2047 = up, 2048..4095 = down).
// LDS layout (dynamic): sRe[128*LDW] | sIm[128*LDW] | sAcc[2] | sPiv[1]
__global__ __launch_bounds__(256)
void slater_lu_kernel(const int* __restrict__ x,
                      const float2* __restrict__ Uup,
                      const float2* __restrict__ Udn,
                      float2* __restrict__ ws)
{
    extern __shared__ float smem[];
    float* sRe  = smem;
    float* sIm  = smem + N * LDW;
    float* sAcc = smem + 2 * N * LDW;       // [0] = sum log|pivot|, [1] = sum angle
    int*   sPiv = (int*)(sAcc + 2);

    const int tid = threadIdx.x;
    const int b   = blockIdx.x;
    const bool up = (b < NBATCH);
    const int batch = up ? b : b - NBATCH;
    const int* xrow = x + batch * 256 + (up ? 0 : N);
    const float2* U = up ? Uup : Udn;

    if (tid == 0) { sAcc[0] = 0.0f; sAcc[1] = 0.0f; }

    // ---- Gather: A[r][c] = U[x[r]][c], deinterleaved re/im into LDS ----
    for (int idx = tid; idx < N * N; idx += 256) {
        int r = idx >> 7, c = idx & (N - 1);
        float2 v = U[(size_t)xrow[r] * N + c];
        sRe[r * LDW + c] = v.x;
        sIm[r * LDW + c] = v.y;
    }

    // ---- Blocked LU with partial pivoting, block = 16 ----
    for (int k0 = 0; k0 < N; k0 += 16) {
        // ===== Panel factorization: columns k0 .. k0+15 =====
        for (int j = k0; j < k0 + 16; ++j) {
            __syncthreads();
            // Pivot search over rows j..127 of column j (wave 0 only)
            if (tid < 32) {
                float best = -1.0f; int brow = j;
                for (int i = j + tid; i < N; i += 32) {
                    float re = sRe[i * LDW + j], im = sIm[i * LDW + j];
                    float m = re * re + im * im;
                    if (m > best) { best = m; brow = i; }
                }
                #pragma unroll
                for (int off = 16; off > 0; off >>= 1) {
                    float ob = __shfl_xor(best, off, 32);
                    int   orow = __shfl_xor(brow, off, 32);
                    if (ob > best) { best = ob; brow = orow; }
                }
                if (tid == 0) {
                    sPiv[0] = brow;
                    float pr = sRe[brow * LDW + j], pim = sIm[brow * LDW + j];
                    sAcc[0] += 0.5f * logf(best);          // log |pivot|
                    float ang = atan2f(pim, pr);
                    if (brow != j) ang += PI_F;            // row-swap sign flip
                    sAcc[1] += ang;
                }
            }
            __syncthreads();
            // Full-row swap (both planes), 128 threads, one column each
            int piv = sPiv[0];
            if (piv != j && tid < N) {
                float tr = sRe[j * LDW + tid];
                sRe[j * LDW + tid] = sRe[piv * LDW + tid];
                sRe[piv * LDW + tid] = tr;
                float ti_ = sIm[j * LDW + tid];
                sIm[j * LDW + tid] = sIm[piv * LDW + tid];
                sIm[piv * LDW + tid] = ti_;
            }
            __syncthreads();
            // Scale column j below diagonal + rank-1 update within the panel
            int i = j + 1 + tid;
            if (i < N) {
                float pr = sRe[j * LDW + j], pim = sIm[j * LDW + j];
                float inv = 1.0f / (pr * pr + pim * pim);
                float ipr = pr * inv, ipi = -pim * inv;    // 1/p = conj(p)/|p|^2
                float ar = sRe[i * LDW + j], ai = sIm[i * LDW + j];
                float lr = ar * ipr - ai * ipi;
                float li = ar * ipi + ai * ipr;
                sRe[i * LDW + j] = lr;
                sIm[i * LDW + j] = li;
                for (int c = j + 1; c < k0 + 16; ++c) {
                    float ur = sRe[j * LDW + c], ui = sIm[j * LDW + c];
                    sRe[i * LDW + c] -= lr * ur - li * ui;
                    sIm[i * LDW + c] -= lr * ui + li * ur;
                }
            }
        }
        __syncthreads();

        // ===== TRSM: U12 = L11^{-1} A12 (unit lower tri, fwd substitution) =====
        {
            int c = k0 + 16 + tid;
            if (c < N) {
                for (int r = k0 + 1; r < k0 + 16; ++r) {
                    float accr = sRe[r * LDW + c], acci = sIm[r * LDW + c];
                    for (int s = k0; s < r; ++s) {
                        float lr = sRe[r * LDW + s], li = sIm[r * LDW + s];
                        float ur = sRe[s * LDW + c], ui = sIm[s * LDW + c];
                        accr -= lr * ur - li * ui;
                        acci -= lr * ui + li * ur;
                    }
                    sRe[r * LDW + c] = accr;
                    sIm[r * LDW + c] = acci;
                }
            }
        }
        __syncthreads();

        // ===== Trailing update A22 -= L21 * U12 via WMMA f32 16x16x4 =====
        // Complex GEMM-subtract as 4 real WMMA accumulations per k-slice:
        //   Cr += (-Ar)Br + (Ai)Bi ;  Ci += (-Ar)Bi + (-Ai)Br
        int kb = k0 >> 4;
        int nb = 7 - kb;                 // number of 16-wide trailing blocks
        if (nb > 0) {
            const int w    = tid >> 5;   // wave id (8 waves)
            const int lane = tid & 31;
            const int m15  = lane & 15;
            const int hi   = lane >> 4;  // half-wave select
            for (int t = w; t < nb * nb; t += 8) {
                int ti = t / nb, tj = t - ti * nb;
                int r0 = (kb + 1 + ti) * 16;
                int c0 = (kb + 1 + tj) * 16;
                // Load C accumulators (16x16 f32 layout: VGPR r -> M = r + 8*hi, N = m15)
                v8f cr, ci;
                #pragma unroll
                for (int r = 0; r < 8; ++r) {
                    int row = r0 + r + 8 * hi;
                    cr[r] = sRe[row * LDW + c0 + m15];
                    ci[r] = sIm[row * LDW + c0 + m15];
                }
                #pragma unroll
                for (int ks = 0; ks < 4; ++ks) {
                    int k  = k0 + 4 * ks;
                    int ka = k + 2 * hi;     // A: lane holds K = ka, ka+1 for row m15
                    v2f ar, ai, br, bi;
                    ar.x = sRe[(r0 + m15) * LDW + ka];
                    ar.y = sRe[(r0 + m15) * LDW + ka + 1];
                    ai.x = sIm[(r0 + m15) * LDW + ka];
                    ai.y = sIm[(r0 + m15) * LDW + ka + 1];
                    // B: lane holds rows K = ka, ka+1 at column c0+m15
                    br.x = sRe[ka * LDW + c0 + m15];
                    br.y = sRe[(ka + 1) * LDW + c0 + m15];
                    bi.x = sIm[ka * LDW + c0 + m15];
                    bi.y = sIm[(ka + 1) * LDW + c0 + m15];
                    v2f arn = -ar;
                    v2f ain = -ai;
                    cr = __builtin_amdgcn_wmma_f32_16x16x4_f32(false, arn, false, br,
                                                               (short)0, cr, false, false);
                    cr = __builtin_amdgcn_wmma_f32_16x16x4_f32(false, ai,  false, bi,
                                                               (short)0, cr, false, false);
                    ci = __builtin_amdgcn_wmma_f32_16x16x4_f32(false, arn, false, bi,
                                                               (short)0, ci, false, false);
                    ci = __builtin_amdgcn_wmma_f32_16x16x4_f32(false, ain, false, br,
                                                               (short)0, ci, false, false);
                }
                #pragma unroll
                for (int r = 0; r < 8; ++r) {
                    int row = r0 + r + 8 * hi;
                    sRe[row * LDW + c0 + m15] = cr[r];
                    sIm[row * LDW + c0 + m15] = ci[r];
                }
            }
        }
        // next panel's first __syncthreads() orders GEMM writes vs pivot reads
    }

    __syncthreads();
    if (tid == 0) {
        ws[b] = make_float2(sAcc[0], sAcc[1]);
    }
}

// Combine up/down halves: out = (l_up + l_dn) + i * wrap(a_up + a_dn)
__global__ __launch_bounds__(256)
void slater_combine_kernel(const float2* __restrict__ ws, float2* __restrict__ out)
{
    int i = blockIdx.x * blockDim.x + threadIdx.x;
    if (i < NBATCH) {
        float2 u = ws[i];
        float2 d = ws[i + NBATCH];
        float re = u.x + d.x;
        float an = u.y + d.y;
        an = fmodf(an, TWO_PI_F);
        if (an > PI_F)        an -= TWO_PI_F;
        else if (an <= -PI_F) an += TWO_PI_F;
        out[i] = make_float2(re, an);
    }
}

extern "C" void kernel_launch(void* const* d_in, const int* in_sizes, int n_in,
                              void* d_out, int out_size, void* d_ws, size_t ws_size,
                              hipStream_t stream)
{
    (void)in_sizes; (void)n_in; (void)out_size; (void)ws_size;
    const int*    x   = (const int*)d_in[0];
    const float2* Uup = (const float2*)d_in[1];   // complex64 interleaved
    const float2* Udn = (const float2*)d_in[2];   // complex64 interleaved
    float2* ws  = (float2*)d_ws;                  // 4096 (log|det|, angle) pairs
    float2* out = (float2*)d_out;                 // 2048 complex64 results

    size_t shmem = (size_t)(2 * N * LDW + 8) * sizeof(float);  // ~135 KB of 320 KB WGP LDS
    slater_lu_kernel<<<2 * NBATCH, 256, shmem, stream>>>(x, Uup, Udn, ws);
    slater_combine_kernel<<<(NBATCH + 255) / 256, 256, 0, stream>>>(ws, out);
}